// CTCLoss_70875550319114
// MI455X (gfx1250) — compile-verified
//
#include <hip/hip_runtime.h>
#include <hip/hip_bf16.h>
#include <stdint.h>

// CTC forward (negative log-likelihood, mean reduction, zero_infinity) for
// B=32, T=2000, V=1024, S=256 on gfx1250.
//
// Strategy: one workgroup per batch sample; 544 threads (17 wave32) = one
// thread per extended-label cell (L=513, rounded up). Alpha lives in LDS,
// ping-pong buffered -> exactly one barrier per time step. The per-step
// emission row log_probs[b,t,:] (4KB) is DMA'd into LDS by the Tensor Data
// Mover, double-buffered: issue row t+1 at the top of step t, wait on
// TENSORcnt at the bottom -> global traffic fully off the critical path.

#define NEGV (-1e30f)

typedef __attribute__((ext_vector_type(4))) unsigned int v4u;
typedef __attribute__((ext_vector_type(4))) int  v4i;
typedef __attribute__((ext_vector_type(8))) int  v8i;

#ifndef __has_builtin
#define __has_builtin(x) 0
#endif

#if __has_builtin(__builtin_amdgcn_tensor_load_to_lds) && \
    __has_builtin(__builtin_amdgcn_s_wait_tensorcnt)
#define CTC_USE_TDM 1
#else
#define CTC_USE_TDM 0
#endif

__device__ __forceinline__ unsigned lds_byte_offset(const void* p) {
  // gfx1250 flat shared addresses: ADDR[31:0] is the LDS byte offset
  // (aperture bits live in ADDR[63:32]).
  return (unsigned)(uintptr_t)p;
}

template <int VMAX, int LMAX>
__global__ __launch_bounds__(LMAX) void ctc_alpha_kernel(
    const float* __restrict__ log_probs,   // (B,T,V) log-softmax
    const int*   __restrict__ targets,     // (B,S)
    const int*   __restrict__ input_lengths,   // (B,)
    const int*   __restrict__ target_lengths,  // (B,)
    float*       __restrict__ per_sample,      // (B,) nll / target_len
    int T, int V, int S)
{
  __shared__ float ebuf[2][VMAX];   // double-buffered emission rows
  __shared__ float abuf[2][LMAX];   // ping-pong alpha

  const int b  = blockIdx.x;
  const int l  = threadIdx.x;
  const int L  = 2 * S + 1;
  const int Tb = input_lengths[b];
  const bool wave0 = (threadIdx.x < 32);

  // Extended label ext[l] (blank=0 at even l, target at odd l) and skip flag:
  // skip transition l-2 -> l allowed iff ext[l] != blank && ext[l] != ext[l-2].
  int  myext  = 0;
  bool myskip = false;
  if (l < L && (l & 1)) {
    int s  = (l - 1) >> 1;
    myext  = targets[b * S + s];
    myskip = (l >= 3) && (myext != targets[b * S + s - 1]);
  }

  const float* row0 = log_probs + (size_t)b * (size_t)T * (size_t)V;

#if CTC_USE_TDM
  // ---- Tensor DMA descriptor, group 1 (constant across the scan) ----
  v8i g1;
  g1[0] = 0x00020000;                          // data_size = 2 (4-byte elems)
  g1[1] = (V & 0xFFFF) << 16;                  // tensor_dim0[15:0]
  g1[2] = (int)(((unsigned)V >> 16) | (1u << 16)); // tensor_dim0[31:16], tensor_dim1=1
  g1[3] = (V & 0xFFFF) << 16;                  // tile_dim0 = V  (bits 127:112)
  g1[4] = 0;                                   // tile_dim1 = tile_dim2 = 0
  g1[5] = V;                                   // tensor_dim0_stride[31:0]
  g1[6] = 0;
  g1[7] = 0;
  const v4i gz4 = {0, 0, 0, 0};                // groups 2/3 unused (<=2D)
  const v8i gz8 = {0, 0, 0, 0, 0, 0, 0, 0};    // trailing group block (unused)

  auto issue = [&](int t, int buf) {
    unsigned long long ga =
        (unsigned long long)(uintptr_t)(row0 + (size_t)t * (size_t)V);
    v4u g0;
    g0[0] = 1u;                                      // count=1, user mode
    g0[1] = lds_byte_offset(&ebuf[buf][0]);          // lds_addr
    g0[2] = (unsigned)(ga & 0xFFFFFFFFull);          // global_addr[31:0]
    g0[3] = (unsigned)((ga >> 32) & 0x01FFFFFFull)   // global_addr[56:32]
            | 0x80000000u;                           // type = 2 ("image")
    __builtin_amdgcn_tensor_load_to_lds(g0, g1, gz4, gz4, gz8, 0);
  };

  if (wave0) {
    issue(0, 0);
    if (Tb > 1) issue(1, 1);
    __builtin_amdgcn_s_wait_tensorcnt(0);
  }
#else
  for (int i = threadIdx.x; i < V; i += blockDim.x) ebuf[0][i] = row0[i];
  if (Tb > 1)
    for (int i = threadIdx.x; i < V; i += blockDim.x) ebuf[1][i] = row0[V + i];
#endif
  __syncthreads();

  // alpha_0[l] = emit(0,l) for l<2 else NEG
  if (l < L) abuf[0][l] = (l < 2) ? ebuf[0][myext] : NEGV;
  __syncthreads();

  int cur = 0;
  for (int t = 1; t < Tb; ++t) {
#if CTC_USE_TDM
    if (wave0 && (t + 1 < Tb)) issue(t + 1, (t + 1) & 1);  // prefetch next row
#endif
    if (l < L) {
      const float* ap = abuf[cur];
      float a  = ap[l];
      float p1 = (l >= 1) ? ap[l - 1] : NEGV;
      float p2 = myskip   ? ap[l - 2] : NEGV;
      float m  = fmaxf(a, fmaxf(p1, p2));
      float lse = m + __logf(__expf(a - m) + __expf(p1 - m) + __expf(p2 - m));
      abuf[cur ^ 1][l] = lse + ebuf[t & 1][myext];   // LDS gather of emission
    }
#if CTC_USE_TDM
    if (wave0) __builtin_amdgcn_s_wait_tensorcnt(0); // row t+1 landed
#else
    if (t + 1 < Tb) {
      const float* rowp = row0 + (size_t)(t + 1) * (size_t)V;
      for (int i = threadIdx.x; i < V; i += blockDim.x)
        ebuf[(t + 1) & 1][i] = rowp[i];
    }
#endif
    __syncthreads();   // publishes alpha_t and emission row t+1
    cur ^= 1;
  }

  if (threadIdx.x == 0) {
    const int    tl = target_lengths[b];
    const float* af = abuf[cur];                 // alpha at t = Tb-1
    float e1 = af[2 * tl];
    float e2 = af[2 * tl - 1];
    float m  = fmaxf(e1, e2);
    float ll = m + __logf(__expf(e1 - m) + __expf(e2 - m));
    float nll = -ll;
    if (nll > 0.5e30f) nll = 0.0f;               // zero_infinity
    per_sample[b] = nll / (float)tl;             // per-sample mean prep
  }
}

__global__ void ctc_mean_kernel(const float* __restrict__ per_sample,
                                float* __restrict__ out, int B) {
  if (blockIdx.x == 0 && threadIdx.x == 0) {
    float s = 0.f;
    for (int i = 0; i < B; ++i) s += per_sample[i];
    out[0] = s / (float)B;
  }
}

extern "C" void kernel_launch(void* const* d_in, const int* in_sizes, int n_in,
                              void* d_out, int out_size, void* d_ws, size_t ws_size,
                              hipStream_t stream) {
  const float* log_probs      = (const float*)d_in[0];
  const int*   targets        = (const int*)d_in[1];
  const int*   input_lengths  = (const int*)d_in[2];
  const int*   target_lengths = (const int*)d_in[3];

  const int B = in_sizes[2];            // 32
  const int S = in_sizes[1] / B;        // 256
  const int V = 1024;                   // vocab (fixed by reference)
  const int T = in_sizes[0] / (B * V);  // 2000

  float* per_sample = (float*)d_ws;     // B floats of scratch

  // L = 2*S+1 = 513 -> 544 threads (17 wave32)
  ctc_alpha_kernel<1024, 544><<<dim3(B), dim3(544), 0, stream>>>(
      log_probs, targets, input_lengths, target_lengths, per_sample, T, V, S);

  ctc_mean_kernel<<<dim3(1), dim3(32), 0, stream>>>(per_sample, (float*)d_out, B);
}